// TestHeteroGNN_19043884990820
// MI455X (gfx1250) — compile-verified
//
#include <hip/hip_runtime.h>
#include <hip/hip_bf16.h>

typedef __attribute__((ext_vector_type(2))) float v2f;
typedef __attribute__((ext_vector_type(8))) float v8f;

#define N_NODES 50000
#define N_EDGES 800000
#define DIM     128

// ---------------------------------------------------------------------------
// Kernel 1: zero the workspace (agg sums, degree counts, column sums).
// ---------------------------------------------------------------------------
__global__ void zero_ws_kernel(float* __restrict__ p, size_t n) {
  size_t i = (size_t)blockIdx.x * blockDim.x + threadIdx.x;
  size_t stride = (size_t)gridDim.x * blockDim.x;
  for (; i < n; i += stride) p[i] = 0.0f;
}

// ---------------------------------------------------------------------------
// Kernel 2: edge scatter. One wave per edge; each of the 32 lanes moves a
// float4 (4 of the 128 features) with global f32 atomics into agg[dst].
// agg (25.6 MB) and x (25.6 MB) are L2-resident on MI455X (192 MB L2).
// ---------------------------------------------------------------------------
__global__ void scatter_accum_kernel(const float* __restrict__ x,
                                     const long long* __restrict__ ei, // [2,E]
                                     float* __restrict__ agg,
                                     float* __restrict__ cnt,
                                     int nedges) {
  int wave = (int)((blockIdx.x * blockDim.x + threadIdx.x) >> 5);
  int lane = threadIdx.x & 31;
  if (wave >= nedges) return;               // wave-uniform branch
  long long src = ei[wave];                 // row 0: sources
  long long dst = ei[(size_t)nedges + wave];// row 1: destinations
  const float4* xr = (const float4*)(x + (size_t)src * DIM);
  float4 v = xr[lane];
  float* ar = agg + (size_t)dst * DIM + (size_t)lane * 4;
  atomicAdd(ar + 0, v.x);
  atomicAdd(ar + 1, v.y);
  atomicAdd(ar + 2, v.z);
  atomicAdd(ar + 3, v.w);
  if (lane == 0) atomicAdd(cnt + dst, 1.0f);
}

// ---------------------------------------------------------------------------
// Kernel 3: fused SAGE GEMM + bias + ReLU + column reduction, using
// V_WMMA_F32_16X16X4_F32 (exact f32 math). Each wave owns one 16-node tile
// and computes   out[16,128] = (agg/deg) @ Wl + x @ Wr + bl
// as 8 column tiles of 16x16, K=128 in 32 steps of 4.
//
// f32 WMMA fragment layouts (ISA 7.12.2):
//   A (16x4):  lane L holds A[L&15][2*(L>>4) + {0,1}]        -> v2f
//   B (4x16):  lane L holds B[2*(L>>4) + {0,1}][L&15]        -> v2f
//   C (16x16): VGPR r on lane L = C[r + 8*(L>>4)][L&15]      -> v8f
// ---------------------------------------------------------------------------
__global__ void sage_wmma_reduce_kernel(const float* __restrict__ x,
                                        const float* __restrict__ agg,
                                        const float* __restrict__ cnt,
                                        const float* __restrict__ Wl, // [128,128]
                                        const float* __restrict__ bl, // [128]
                                        const float* __restrict__ Wr, // [128,128]
                                        float* __restrict__ colsum,   // [128]
                                        int nnodes) {
  __shared__ float lsum[DIM];
  int t = threadIdx.x;
  if (t < DIM) lsum[t] = 0.0f;
  __syncthreads();

  int lane = t & 31;
  int tile = blockIdx.x * (blockDim.x >> 5) + (t >> 5); // 16-node tile index
  int ntiles = nnodes >> 4;                             // 50000/16 = 3125

  if (tile < ntiles) {                                  // wave-uniform; EXEC all-1s
    int m  = lane & 15;                                 // A row / B,C column
    int kh = lane >> 4;                                 // K-half select
    int node = tile * 16 + m;
    float inv = 1.0f / fmaxf(cnt[node], 1.0f);          // mean; 0 if isolated
    const float* aggRow = agg + (size_t)node * DIM;
    const float* xRow   = x   + (size_t)node * DIM;

    for (int nt = 0; nt < 8; ++nt) {
      int col = nt * 16 + m;
      v8f c = {};
#pragma unroll 4
      for (int kk = 0; kk < 32; ++kk) {
        int k = kk * 4 + kh * 2;
        v2f aA, aX, bL, bR;
        aA.x = aggRow[k] * inv;  aA.y = aggRow[k + 1] * inv;
        aX.x = xRow[k];          aX.y = xRow[k + 1];
        bL.x = Wl[(size_t)k * DIM + col];  bL.y = Wl[(size_t)(k + 1) * DIM + col];
        bR.x = Wr[(size_t)k * DIM + col];  bR.y = Wr[(size_t)(k + 1) * DIM + col];
        c = __builtin_amdgcn_wmma_f32_16x16x4_f32(false, aA, false, bL,
                                                  (short)0, c, false, false);
        c = __builtin_amdgcn_wmma_f32_16x16x4_f32(false, aX, false, bR,
                                                  (short)0, c, false, false);
      }
      // bias + ReLU + sum over this lane's 8 node-rows of column `col`
      float b = bl[col];
      float s = 0.0f;
#pragma unroll
      for (int r = 0; r < 8; ++r) s += fmaxf(c[r] + b, 0.0f);
      atomicAdd(&lsum[col], s);   // ds_add_f32; lane and lane^16 share a column
    }
  }

  __syncthreads();
  if (t < DIM) atomicAdd(&colsum[t], lsum[t]);  // one global atomic per col/block
}

// ---------------------------------------------------------------------------
// Kernel 4: out = (colsum/N) . W_out + b_out   (single 128-thread block)
// ---------------------------------------------------------------------------
__global__ void final_dot_kernel(const float* __restrict__ colsum,
                                 const float* __restrict__ Wout,
                                 const float* __restrict__ bout,
                                 float* __restrict__ out, float invN) {
  __shared__ float red[DIM];
  int t = threadIdx.x;
  red[t] = colsum[t] * invN * Wout[t];
  __syncthreads();
  for (int s = 64; s > 0; s >>= 1) {
    if (t < s) red[t] += red[t + s];
    __syncthreads();
  }
  if (t == 0) out[0] = red[0] + bout[0];
}

// ---------------------------------------------------------------------------
// Host launcher. Input order follows setup_inputs():
//  0 x_ligand  1 x_protein  2 ei_lp  3 ei_pp  4 ei_ll
//  5 Wl_lp 6 bl_lp 7 Wr_lp  8 Wl_pp 9 bl_pp 10 Wr_pp
// 11 Wl_ll 12 bl_ll 13 Wr_ll 14 W_out 15 b_out
// The output depends only on the ligand branch; the protein branch is dead
// code in the reference and is skipped.
// ---------------------------------------------------------------------------
extern "C" void kernel_launch(void* const* d_in, const int* in_sizes, int n_in,
                              void* d_out, int out_size, void* d_ws, size_t ws_size,
                              hipStream_t stream) {
  (void)in_sizes; (void)n_in; (void)out_size; (void)ws_size;
  const float*     x_lig = (const float*)d_in[0];
  const long long* ei_ll = (const long long*)d_in[4];   // int64 indices
  const float*     Wl    = (const float*)d_in[11];
  const float*     bl    = (const float*)d_in[12];
  const float*     Wr    = (const float*)d_in[13];
  const float*     Wout  = (const float*)d_in[14];
  const float*     bout  = (const float*)d_in[15];
  float*           out   = (float*)d_out;

  float* agg    = (float*)d_ws;                         // N*128 f32 = 25.6 MB
  float* cnt    = agg + (size_t)N_NODES * DIM;          // N f32
  float* colsum = cnt + N_NODES;                        // 128 f32
  size_t zcount = (size_t)N_NODES * DIM + N_NODES + DIM;

  zero_ws_kernel<<<2048, 256, 0, stream>>>((float*)d_ws, zcount);

  // 8 waves/block, one wave per edge: 800000/8 = 100000 blocks exactly.
  scatter_accum_kernel<<<N_EDGES / 8, 256, 0, stream>>>(x_lig, ei_ll, agg, cnt,
                                                        N_EDGES);

  // 3125 node tiles, 8 waves/block -> 391 blocks.
  int ntiles = N_NODES / 16;
  int blocks = (ntiles + 7) / 8;
  sage_wmma_reduce_kernel<<<blocks, 256, 0, stream>>>(x_lig, agg, cnt,
                                                      Wl, bl, Wr, colsum,
                                                      N_NODES);

  final_dot_kernel<<<1, 128, 0, stream>>>(colsum, Wout, bout, out,
                                          1.0f / (float)N_NODES);
}